// MaxwellDemonFilter_88484916232368
// MI455X (gfx1250) — compile-verified
//
#include <hip/hip_runtime.h>
#include <math.h>

// N=50000, E=800000, DIM=128, H=8, HD=16, HID=512
typedef float v2f __attribute__((ext_vector_type(2)));
typedef float v8f __attribute__((ext_vector_type(8)));

__device__ __forceinline__ v8f wmma_f32(v2f a, v2f b, v8f c) {
  // D = A(16x4) * B(4x16) + C(16x16), full fp32
  return __builtin_amdgcn_wmma_f32_16x16x4_f32(false, a, false, b, (short)0, c,
                                               false, false);
}

// ---------------- fill ----------------
__global__ void fill_kernel(float* p, float v, long n) {
  long i = (long)blockIdx.x * blockDim.x + threadIdx.x;
  long stride = (long)gridDim.x * blockDim.x;
  for (; i < n; i += stride) p[i] = v;
}

// ---------------- h = x @ fc_w^T + fc_b ----------------
__global__ __launch_bounds__(128) void fc_gemm_kernel(
    const float* __restrict__ x, const float* __restrict__ w,
    const float* __restrict__ bias, float* __restrict__ h, int N) {
  __shared__ float xs[16 * 132];
  const int tid = threadIdx.x;
  const int node0 = blockIdx.x * 16;
  // stage 16x128 A-tile into LDS (coalesced)
  for (int i = 0; i < 16; ++i) {
    int flat = i * 128 + tid;
    int row = flat >> 7, col = flat & 127;
    int n = node0 + row;
    xs[row * 132 + col] = (n < N) ? x[(size_t)n * 128 + col] : 0.0f;
  }
  __syncthreads();
  const int lane = tid & 31, wave = tid >> 5;
  const int m = lane & 15;
  const int koff = (lane >> 4) * 2;
  const int rowoff = (lane >> 4) * 8;
  const int col0 = wave * 32;
  v8f acc[2] = {};
  const float* w0 = w + (size_t)(col0 + m) * 128;
  const float* w1p = w + (size_t)(col0 + 16 + m) * 128;
  for (int kk = 0; kk < 128; kk += 4) {
    v2f a;
    a.x = xs[m * 132 + kk + koff];
    a.y = xs[m * 132 + kk + koff + 1];
    v2f b0 = *(const v2f*)(w0 + kk + koff);
    v2f b1 = *(const v2f*)(w1p + kk + koff);
    acc[0] = wmma_f32(a, b0, acc[0]);
    acc[1] = wmma_f32(a, b1, acc[1]);
  }
#pragma unroll
  for (int t = 0; t < 2; ++t) {
    int n = col0 + t * 16 + m;
    float bv = bias[n];
#pragma unroll
    for (int r = 0; r < 8; ++r) {
      int node = node0 + r + rowoff;
      if (node < N) h[(size_t)node * 128 + n] = acc[t][r] + bv;
    }
  }
}

// ---------------- su = h@au_w^T+au_b, sv = h@av_w^T (wave per node) --------
__global__ __launch_bounds__(256) void susv_kernel(
    const float* __restrict__ h, const float* __restrict__ au_w,
    const float* __restrict__ au_b, const float* __restrict__ av_w,
    float* __restrict__ su, float* __restrict__ sv, int N) {
  int node = blockIdx.x * 8 + (threadIdx.x >> 5);
  if (node >= N) return;
  int lane = threadIdx.x & 31;
  float hv[4];
#pragma unroll
  for (int j = 0; j < 4; ++j) hv[j] = h[(size_t)node * 128 + lane + 32 * j];
#pragma unroll
  for (int hh = 0; hh < 8; ++hh) {
    float s = 0.f, v = 0.f;
#pragma unroll
    for (int j = 0; j < 4; ++j) {
      s += hv[j] * au_w[hh * 128 + lane + 32 * j];
      v += hv[j] * av_w[hh * 128 + lane + 32 * j];
    }
#pragma unroll
    for (int off = 16; off; off >>= 1) {
      s += __shfl_xor(s, off);
      v += __shfl_xor(v, off);
    }
    if (lane == 0) {
      su[(size_t)node * 8 + hh] = s + au_b[hh];
      sv[(size_t)node * 8 + hh] = v;
    }
  }
}

// ---------------- ordered-float encoding for atomic max ----------------
__device__ __forceinline__ unsigned enc_f(float f) {
  unsigned u = __float_as_uint(f);
  return (f >= 0.0f) ? (u | 0x80000000u) : ~u;
}
__device__ __forceinline__ float dec_f(unsigned e) {
  unsigned u = (e & 0x80000000u) ? (e & 0x7FFFFFFFu) : ~e;
  return __uint_as_float(u);
}

// ---------------- leaky scores + segment max + degrees ----------------
__global__ void edge_score_kernel(
    const float* __restrict__ su, const float* __restrict__ sv,
    const int* __restrict__ src, const int* __restrict__ dst,
    float* __restrict__ scores, unsigned* __restrict__ menc,
    float* __restrict__ deg_in, float* __restrict__ deg_out, int E) {
  int e = blockIdx.x * blockDim.x + threadIdx.x;
  if (e >= E) return;
  int s = src[e], d = dst[e];
  atomicAdd(&deg_out[s], 1.0f);
  atomicAdd(&deg_in[d], 1.0f);
#pragma unroll
  for (int hh = 0; hh < 8; ++hh) {
    float sc = su[(size_t)s * 8 + hh] + sv[(size_t)d * 8 + hh];
    sc = (sc >= 0.f) ? sc : 0.2f * sc;
    scores[(size_t)e * 8 + hh] = sc;
    atomicMax(&menc[(size_t)d * 8 + hh], enc_f(sc));
  }
}

// ---------------- exp(s - m[dst]) + segment sum ----------------
__global__ void edge_exp_kernel(float* __restrict__ scores,
                                const unsigned* __restrict__ menc,
                                const int* __restrict__ dst,
                                float* __restrict__ den, int E) {
  int e = blockIdx.x * blockDim.x + threadIdx.x;
  if (e >= E) return;
  int d = dst[e];
#pragma unroll
  for (int hh = 0; hh < 8; ++hh) {
    float mv = dec_f(menc[(size_t)d * 8 + hh]);
    float ex = expf(scores[(size_t)e * 8 + hh] - mv);
    scores[(size_t)e * 8 + hh] = ex;
    atomicAdd(&den[(size_t)d * 8 + hh], ex);
  }
}

// ---------------- fused message scatter: msg, msg2(sum), msg3 ----------------
// wave per edge; lane handles 4 contiguous feature dims; head = dim & 7
__global__ __launch_bounds__(256) void edge_msg_kernel(
    const float* __restrict__ h, const float* __restrict__ ex,
    const float* __restrict__ den, const int* __restrict__ src,
    const int* __restrict__ dst, const float* __restrict__ deg_out,
    float* __restrict__ msg, float* __restrict__ msg2,
    float* __restrict__ msg3, int E) {
  int e = blockIdx.x * 8 + (threadIdx.x >> 5);
  if (e >= E) return;
  int lane = threadIdx.x & 31;
  int s = src[e], d = dst[e];
  float norm = 1.0f / sqrtf(deg_out[s] * deg_out[d]);
  int half = lane & 1;  // dims lane*4..lane*4+3 map to heads 4*half..4*half+3
  float4 ex4 = ((const float4*)(ex + (size_t)e * 8))[half];
  float4 dn4 = ((const float4*)(den + (size_t)d * 8))[half];
  float4 p;
  p.x = ex4.x / dn4.x; p.y = ex4.y / dn4.y;
  p.z = ex4.z / dn4.z; p.w = ex4.w / dn4.w;
  float4 hv = ((const float4*)(h + (size_t)s * 128))[lane];
  size_t dbase = (size_t)d * 128 + lane * 4;
  atomicAdd(&msg[dbase + 0], hv.x * p.x);
  atomicAdd(&msg[dbase + 1], hv.y * p.y);
  atomicAdd(&msg[dbase + 2], hv.z * p.z);
  atomicAdd(&msg[dbase + 3], hv.w * p.w);
  atomicAdd(&msg2[dbase + 0], hv.x);
  atomicAdd(&msg2[dbase + 1], hv.y);
  atomicAdd(&msg2[dbase + 2], hv.z);
  atomicAdd(&msg2[dbase + 3], hv.w);
  atomicAdd(&msg3[dbase + 0], hv.x * norm);
  atomicAdd(&msg3[dbase + 1], hv.y * norm);
  atomicAdd(&msg3[dbase + 2], hv.z * norm);
  atomicAdd(&msg3[dbase + 3], hv.w * norm);
}

// ---------------- fused FFN: out = gelu(cat@w1^T+b1)@w2^T + b2 ----------------
__global__ __launch_bounds__(128) void ffn_kernel(
    const float* __restrict__ h, const float* __restrict__ msg,
    const float* __restrict__ msg2s, const float* __restrict__ msg3,
    const float* __restrict__ deg_in, const float* __restrict__ w1,
    const float* __restrict__ b1, const float* __restrict__ w2,
    const float* __restrict__ b2, float* __restrict__ out, int N) {
  __shared__ float tile[16 * 516];  // 16 x 512 (+pad): cat tile, then f tile
  const int tid = threadIdx.x;
  const int node0 = blockIdx.x * 16;
  // build cat tile = [h | msg | msg2sum/deg | msg3]
  for (int i = 0; i < 64; ++i) {
    int flat = i * 128 + tid;
    int row = flat >> 9, col = flat & 511;
    int n = node0 + row;
    float v = 0.f;
    if (n < N) {
      if (col < 128)      v = h[(size_t)n * 128 + col];
      else if (col < 256) v = msg[(size_t)n * 128 + col - 128];
      else if (col < 384) v = msg2s[(size_t)n * 128 + col - 256] /
                              fmaxf(deg_in[n], 1.0f);
      else                v = msg3[(size_t)n * 128 + col - 384];
    }
    tile[row * 516 + col] = v;
  }
  __syncthreads();
  const int lane = tid & 31, wave = tid >> 5;
  const int m = lane & 15;
  const int koff = (lane >> 4) * 2;
  const int rowoff = (lane >> 4) * 8;
  // stage 1: f = gelu(cat @ w1^T + b1); this wave covers cols [wave*128, +128)
  const int nbase = wave * 128;
  v8f acc[8] = {};
  for (int kk = 0; kk < 512; kk += 4) {
    v2f a;
    a.x = tile[m * 516 + kk + koff];
    a.y = tile[m * 516 + kk + koff + 1];
#pragma unroll
    for (int t = 0; t < 8; ++t) {
      const float* wr = w1 + (size_t)(nbase + t * 16 + m) * 512;
      v2f bb = *(const v2f*)(wr + kk + koff);
      acc[t] = wmma_f32(a, bb, acc[t]);
    }
  }
  __syncthreads();  // all waves done reading cat tile
#pragma unroll
  for (int t = 0; t < 8; ++t) {
    int n = nbase + t * 16 + m;
    float bv = b1[n];
#pragma unroll
    for (int r = 0; r < 8; ++r) {
      float v = acc[t][r] + bv;
      v = 0.5f * v * (1.0f + erff(v * 0.70710678118654752f));  // exact gelu
      tile[(r + rowoff) * 516 + n] = v;
    }
  }
  __syncthreads();
  // stage 2: out = f @ w2^T + b2; this wave covers out cols [wave*32, +32)
  const int obase = wave * 32;
  v8f a2[2] = {};
  for (int kk = 0; kk < 512; kk += 4) {
    v2f a;
    a.x = tile[m * 516 + kk + koff];
    a.y = tile[m * 516 + kk + koff + 1];
#pragma unroll
    for (int t = 0; t < 2; ++t) {
      const float* wr = w2 + (size_t)(obase + t * 16 + m) * 512;
      v2f bb = *(const v2f*)(wr + kk + koff);
      a2[t] = wmma_f32(a, bb, a2[t]);
    }
  }
#pragma unroll
  for (int t = 0; t < 2; ++t) {
    int o = obase + t * 16 + m;
    float bv = b2[o];
#pragma unroll
    for (int r = 0; r < 8; ++r) {
      int node = node0 + r + rowoff;
      if (node < N) out[(size_t)node * 128 + o] = a2[t][r] + bv;
    }
  }
}

extern "C" void kernel_launch(void* const* d_in, const int* in_sizes, int n_in,
                              void* d_out, int out_size, void* d_ws,
                              size_t ws_size, hipStream_t stream) {
  const float* x    = (const float*)d_in[0];
  const int*   src  = (const int*)d_in[1];
  const int*   dst  = (const int*)d_in[2];
  const float* fc_w = (const float*)d_in[3];
  const float* fc_b = (const float*)d_in[4];
  const float* au_w = (const float*)d_in[5];
  const float* au_b = (const float*)d_in[6];
  const float* av_w = (const float*)d_in[7];
  const float* w1   = (const float*)d_in[8];
  const float* b1   = (const float*)d_in[9];
  const float* w2   = (const float*)d_in[10];
  const float* b2   = (const float*)d_in[11];
  const int N = in_sizes[0] / 128;
  const int E = in_sizes[1];

  float* ws = (float*)d_ws;
  float* h      = ws; ws += (size_t)N * 128;
  float* su     = ws; ws += (size_t)N * 8;
  float* sv     = ws; ws += (size_t)N * 8;
  float* scores = ws; ws += (size_t)E * 8;
  float* zbase  = ws;  // everything below must be zeroed each call
  float* den    = ws; ws += (size_t)N * 8;
  unsigned* menc = (unsigned*)ws; ws += (size_t)N * 8;
  float* deg_in  = ws; ws += (size_t)N;
  float* deg_out = ws; ws += (size_t)N;
  float* msg   = ws; ws += (size_t)N * 128;
  float* msg2  = ws; ws += (size_t)N * 128;
  float* msg3  = ws; ws += (size_t)N * 128;
  const long zcount = (long)(ws - zbase);

  fill_kernel<<<(unsigned)((zcount + 255) / 256), 256, 0, stream>>>(
      zbase, 0.0f, zcount);

  const int tiles = (N + 15) / 16;
  fc_gemm_kernel<<<tiles, 128, 0, stream>>>(x, fc_w, fc_b, h, N);
  susv_kernel<<<(N + 7) / 8, 256, 0, stream>>>(h, au_w, au_b, av_w, su, sv, N);
  edge_score_kernel<<<(E + 255) / 256, 256, 0, stream>>>(
      su, sv, src, dst, scores, menc, deg_in, deg_out, E);
  edge_exp_kernel<<<(E + 255) / 256, 256, 0, stream>>>(scores, menc, dst, den,
                                                       E);
  edge_msg_kernel<<<(E + 7) / 8, 256, 0, stream>>>(
      h, scores, den, src, dst, deg_out, msg, msg2, msg3, E);
  ffn_kernel<<<tiles, 128, 0, stream>>>(h, msg, msg2, msg3, deg_in, w1, b1, w2,
                                        b2, (float*)d_out, N);
}